// OnlineSiameseModel_86002425135831
// MI455X (gfx1250) — compile-verified
//
#include <hip/hip_runtime.h>
#include <math.h>
#include <stdint.h>

typedef float    v2f __attribute__((ext_vector_type(2)));
typedef float    v8f __attribute__((ext_vector_type(8)));
typedef uint32_t v4u __attribute__((ext_vector_type(4)));
typedef uint32_t v8u __attribute__((ext_vector_type(8)));

#define N_ROWS  2048
#define F_DIM   16384
#define D_DIM   128
#define ALPHA_C 0.5f
#define SORT_N  2048
#define TPB     512
#define KC      256                 // K-chunk per TDM tile (16 rows x 256 f32 = 16 KB)
#define NCHUNK  (F_DIM / KC)        // 64

// ---------------------------------------------------------------------------
// TDM: issue a tensor_load_to_lds of a 16-row x KC-col f32 tile of `imgs`
// (2D tensor 2048 x 16384, row stride 16384) into LDS at byte offset lds_off.
// D# built per CDNA5 ISA ch.8: group0 = {count=1, lds_addr, global_addr,
// type=2}; group1 = {data_size=4B, tensor_dim0/1, tile_dim0/1, dim0_stride}.
// Groups 2/3 are NULL (<=2D tensor). Tracked with TENSORcnt.
// ---------------------------------------------------------------------------
__device__ __forceinline__ void tdm_load_tile(uint32_t lds_off, uint64_t gaddr)
{
    v4u g0;
    g0[0] = 1u;                                            // count=1, user mode
    g0[1] = lds_off;                                       // LDS dest (bytes)
    g0[2] = (uint32_t)gaddr;                               // global_addr[31:0]
    g0[3] = (uint32_t)((gaddr >> 32) & 0x1FFFFFFu)         // global_addr[56:32]
          | 0x80000000u;                                   // type=2 ("image")

    v8u g1;
    g1[0] = (2u << 16);                                    // data_size=4B; no mcast/pad/iter
    g1[1] = ((uint32_t)(F_DIM & 0xFFFF)) << 16;            // tensor_dim0[15:0]
    g1[2] = ((uint32_t)F_DIM >> 16)
          | (((uint32_t)(N_ROWS & 0xFFFF)) << 16);         // tensor_dim0[31:16] | tensor_dim1[15:0]
    g1[3] = ((uint32_t)N_ROWS >> 16)
          | (((uint32_t)KC) << 16);                        // tensor_dim1[31:16] | tile_dim0=KC
    g1[4] = 16u;                                           // tile_dim1=16 rows, tile_dim2=0
    g1[5] = (uint32_t)F_DIM;                               // tensor_dim0_stride[31:0]
    g1[6] = 0u;                                            // stride[47:32] | dim1_stride lo
    g1[7] = 0u;

    asm volatile("tensor_load_to_lds %0, %1, null, null"
                 :
                 : "s"(g0), "s"(g1)
                 : "memory");
}

// ---------------------------------------------------------------------------
// K1: embs[2048,128] = imgs[2048,16384] @ W[16384,128] + b
// One block = 16 output rows; 8 waves each own a 16x16 tile (wave w -> cols
// w*16..+15). A-tile (16 x KC f32) is DMA'd into LDS by the TDM, double-
// buffered, overlapped with the WMMA loop. B (W, L2-resident) direct global.
// K-loop steps 4 using V_WMMA_F32_16X16X4_F32:
//   A-frag: lanes 0-15 M=0..15 / K pair {0,1}; lanes 16-31 K pair {2,3}
//   B-frag mirrors with N across lanes.
// ---------------------------------------------------------------------------
__global__ __launch_bounds__(256) void gemm_embs(const float* __restrict__ imgs,
                                                 const float* __restrict__ W,
                                                 const float* __restrict__ bias,
                                                 float* __restrict__ embs)
{
    __shared__ float sA[2][16 * KC];                       // 2 x 16 KB

    const int lane = threadIdx.x & 31;
    const int wave = threadIdx.x >> 5;
    const int l15  = lane & 15;
    const int half = lane >> 4;                            // 0: K{0,1}, 1: K{2,3}
    const int m0   = blockIdx.x * 16;
    const int n0   = wave * 16;

    const uint64_t abase = (uint64_t)(uintptr_t)imgs + (uint64_t)m0 * F_DIM * 4u;
    const uint32_t lds0  = (uint32_t)(uintptr_t)&sA[0][0]; // LDS aperture: low 32 bits
    const uint32_t lds1  = (uint32_t)(uintptr_t)&sA[1][0];

    if (wave == 0)
        tdm_load_tile(lds0, abase);                        // prefetch chunk 0

    const float* __restrict__ bcol = W + n0 + l15;

    v8f acc = {};
    for (int c = 0; c < NCHUNK; ++c) {
        if (wave == 0)
            __builtin_amdgcn_s_wait_tensorcnt(0);          // chunk c landed in LDS
        __syncthreads();                                   // visible to all waves
        if (wave == 0 && c + 1 < NCHUNK)                   // async prefetch chunk c+1
            tdm_load_tile((c & 1) ? lds0 : lds1,
                          abase + (uint64_t)(c + 1) * KC * 4u);

        const float* sa    = &sA[c & 1][0] + (size_t)l15 * KC + 2 * half;
        const int    kbase = c * KC;
#pragma unroll 8
        for (int k = 0; k < KC; k += 4) {
            v2f a, b;
            a.x = sa[k];                                   // ds_load_b64
            a.y = sa[k + 1];
            const float* bp = bcol + (size_t)(kbase + k + 2 * half) * D_DIM;
            b.x = bp[0];
            b.y = bp[D_DIM];
            acc = __builtin_amdgcn_wmma_f32_16x16x4_f32(false, a, false, b,
                                                        (short)0, acc, false, false);
        }
        __syncthreads();                                   // buffer reuse fence
    }

    const int   Nc = n0 + l15;
    const float bb = bias[Nc];
#pragma unroll
    for (int r = 0; r < 8; ++r) {
        const int M = m0 + r + 8 * half;                   // C/D layout: VGPR r -> M=r / r+8
        embs[(size_t)M * D_DIM + Nc] = acc[r] + bb;
    }
}

// ---------------------------------------------------------------------------
// K2: sq[i] = sum_k embs[i,k]^2
// ---------------------------------------------------------------------------
__global__ void row_sqnorm(const float* __restrict__ embs, float* __restrict__ sq)
{
    const int r = blockIdx.x * blockDim.x + threadIdx.x;
    if (r < N_ROWS) {
        const float4* p = (const float4*)(embs + (size_t)r * D_DIM);
        float s = 0.f;
#pragma unroll 4
        for (int i = 0; i < D_DIM / 4; ++i) {
            float4 v = p[i];
            s += v.x * v.x + v.y * v.y + v.z * v.z + v.w * v.w;
        }
        sq[r] = s;
    }
}

// ---------------------------------------------------------------------------
// K3: D[i,j] = max(0, sq[i] + sq[j] - 2 * <e_i, e_j>) via WMMA Gram tiles.
// Block tile 32(M) x 64(N): 8 waves in a 2x4 arrangement, K=128 -> 32 WMMAs.
// embs is 1 MB -> fully L2-resident; no staging needed.
// ---------------------------------------------------------------------------
__global__ __launch_bounds__(256) void gram_dist(const float* __restrict__ embs,
                                                 const float* __restrict__ sq,
                                                 float* __restrict__ Dm)
{
    const int lane = threadIdx.x & 31;
    const int wave = threadIdx.x >> 5;
    const int l15  = lane & 15;
    const int half = lane >> 4;
    const int m0   = blockIdx.y * 32 + (wave >> 2) * 16;
    const int n0   = blockIdx.x * 64 + (wave & 3) * 16;

    const float* __restrict__ ap = embs + (size_t)(m0 + l15) * D_DIM + 2 * half;
    const float* __restrict__ bp = embs + (size_t)(n0 + l15) * D_DIM + 2 * half;

    v8f acc = {};
#pragma unroll 4
    for (int k = 0; k < D_DIM; k += 4) {
        v2f a, b;
        a.x = ap[k];
        a.y = ap[k + 1];
        b.x = bp[k];        // B[k][n] = embs[n][k] (embs^T), k contiguous in row
        b.y = bp[k + 1];
        acc = __builtin_amdgcn_wmma_f32_16x16x4_f32(false, a, false, b,
                                                    (short)0, acc, false, false);
    }

    const int   Nc  = n0 + l15;
    const float sqn = sq[Nc];
#pragma unroll
    for (int r = 0; r < 8; ++r) {
        const int   M = m0 + r + 8 * half;
        const float v = sq[M] + sqn - 2.0f * acc[r];
        Dm[(size_t)M * N_ROWS + Nc] = fmaxf(v, 0.0f);
    }
}

// ---------------------------------------------------------------------------
// Deterministic hash standing in for the JAX threefry fallback RNG.
// ---------------------------------------------------------------------------
__device__ __forceinline__ unsigned mix_hash(unsigned a, unsigned b)
{
    unsigned x = a * 0x9E3779B9u + b * 0x85EBCA77u + 0x6C62272Eu;
    x ^= x >> 16; x *= 0x7FEB352Du;
    x ^= x >> 15; x *= 0x846CA68Bu;
    x ^= x >> 16;
    return x;
}

// ---------------------------------------------------------------------------
// K4: one block per anchor row i.
//   - stage D row + masked-negative row in LDS
//   - in-LDS bitonic sort (ascending, +inf on same-label slots)
//   - per column j: upper_bound -> semi-hard idx, hash fallback,
//     masked triplet loss; block-reduce; global atomic accumulate.
// ---------------------------------------------------------------------------
__global__ __launch_bounds__(TPB) void row_loss(const float* __restrict__ Dm,
                                                const int* __restrict__ labels,
                                                float* __restrict__ accum)
{
    __shared__ float s[SORT_N];
    __shared__ float dp[SORT_N];
    __shared__ float rl[TPB];
    __shared__ int   rc[TPB];
    __shared__ int   sh_nneg;

    const int i   = blockIdx.x;
    const int tid = threadIdx.x;
    const int li  = labels[i];
    if (tid == 0) sh_nneg = 0;
    __syncthreads();

    const float* __restrict__ drow = Dm + (size_t)i * SORT_N;
    int cnt = 0;
#pragma unroll
    for (int e = 0; e < SORT_N / TPB; ++e) {
        const int   j    = tid + e * TPB;
        const float d    = drow[j];
        const bool  same = (labels[j] == li);
        dp[j] = d;
        s[j]  = same ? INFINITY : d;
        cnt  += same ? 0 : 1;
    }
    atomicAdd(&sh_nneg, cnt);
    __syncthreads();

    // bitonic sort, ascending
    for (int k = 2; k <= SORT_N; k <<= 1) {
        for (int jj = k >> 1; jj > 0; jj >>= 1) {
#pragma unroll
            for (int e = 0; e < SORT_N / TPB; ++e) {
                const int idx = tid + e * TPB;
                const int l   = idx ^ jj;
                if (l > idx) {
                    const float a   = s[idx];
                    const float b   = s[l];
                    const bool  asc = ((idx & k) == 0);
                    if ((a > b) == asc) { s[idx] = b; s[l] = a; }
                }
            }
            __syncthreads();
        }
    }

    const int nneg  = sh_nneg;
    const int denom = nneg > 1 ? nneg : 1;
    float lsum = 0.f;
    int   lcnt = 0;
#pragma unroll
    for (int e = 0; e < SORT_N / TPB; ++e) {
        const int   j    = tid + e * TPB;
        const bool  same = (labels[j] == li);
        const float dpos = dp[j];
        // searchsorted(side='right'): first index with s[idx] > dpos
        int lo = 0, hi = SORT_N;
        while (lo < hi) {
            const int mid = (lo + hi) >> 1;
            if (s[mid] <= dpos) lo = mid + 1; else hi = mid;
        }
        const bool has = lo < nneg;
        const unsigned rr = mix_hash((unsigned)i, (unsigned)j) & 0x3FFFFFFFu;
        int sel = has ? lo : (int)(rr % (unsigned)denom);
        if (sel > SORT_N - 1) sel = SORT_N - 1;
        const float dneg = s[sel];
        if (same && (i < j)) {
            const float L = dpos - dneg + ALPHA_C;
            if (L > 0.f) lsum += L;
            lcnt += 1;
        }
    }

    rl[tid] = lsum;
    rc[tid] = lcnt;
    __syncthreads();
    for (int st = TPB / 2; st > 0; st >>= 1) {
        if (tid < st) { rl[tid] += rl[tid + st]; rc[tid] += rc[tid + st]; }
        __syncthreads();
    }
    if (tid == 0) {
        atomicAdd(&accum[0], rl[0]);
        atomicAdd(&accum[1], (float)rc[0]);
    }
}

__global__ void zero_accum(float* a)
{
    if (threadIdx.x < 2) a[threadIdx.x] = 0.f;
}

__global__ void finalize(const float* __restrict__ a, float* __restrict__ out)
{
    out[0] = a[0] / fmaxf(a[1], 1.0f);
}

// ---------------------------------------------------------------------------
extern "C" void kernel_launch(void* const* d_in, const int* in_sizes, int n_in,
                              void* d_out, int out_size, void* d_ws, size_t ws_size,
                              hipStream_t stream)
{
    (void)in_sizes; (void)n_in; (void)out_size; (void)ws_size;

    const float* imgs   = (const float*)d_in[0];   // [2048,16384]
    const float* W      = (const float*)d_in[1];   // [16384,128]
    const float* bias   = (const float*)d_in[2];   // [128]
    const int*   labels = (const int*)d_in[3];     // [2048]
    float*       out    = (float*)d_out;           // scalar

    float* embs  = (float*)d_ws;                         // 2048*128
    float* sq    = embs + (size_t)N_ROWS * D_DIM;        // 2048
    float* Dm    = sq + N_ROWS;                          // 2048*2048
    float* accum = Dm + (size_t)N_ROWS * N_ROWS;         // 2

    gemm_embs<<<N_ROWS / 16, 256, 0, stream>>>(imgs, W, bias, embs);
    row_sqnorm<<<N_ROWS / 256, 256, 0, stream>>>(embs, sq);
    gram_dist<<<dim3(N_ROWS / 64, N_ROWS / 32), 256, 0, stream>>>(embs, sq, Dm);
    zero_accum<<<1, 32, 0, stream>>>(accum);
    row_loss<<<N_ROWS, TPB, 0, stream>>>(Dm, labels, accum);
    finalize<<<1, 1, 0, stream>>>(accum, out);
}